// GraphAlphaGenerator_51840255263100
// MI455X (gfx1250) — compile-verified
//
#include <hip/hip_runtime.h>
#include <math.h>

typedef float v2f __attribute__((ext_vector_type(2)));
typedef float v8f __attribute__((ext_vector_type(8)));

#define BB 8
#define LL 8192
#define CC 512
#define HID 16
#define NSPLIT 128
#define LCHUNK (LL / NSPLIT)   // 64

// ---- workspace layout (floats) ----
#define WS_PART   0
#define WS_PART_N (BB * NSPLIT * CC * 3)            // 1572864 (6 MB)
#define WS_HLOCAL (WS_PART + WS_PART_N)
#define WS_Q      (WS_HLOCAL + BB * CC * HID)
#define WS_K      (WS_Q      + BB * CC * HID)
#define WS_HG     (WS_K      + BB * CC * HID)
#define WS_PRIOR  (WS_HG     + BB * CC * HID)
#define WS_MSG    (WS_PRIOR  + CC * CC)

// fp32 WMMA 16x16 tile, K=4 per instruction (exact fp32 math)
__device__ __forceinline__ v8f wmma4(v2f a, v2f b, v8f c) {
  return __builtin_amdgcn_wmma_f32_16x16x4_f32(false, a, false, b, (short)0, c,
                                               false, false);
}

// ------- Kernel A: split-K stats over L, b128 loads (bandwidth kernel) ------
// 128 threads/block; thread t owns channels [4t, 4t+3] via one float4 load
// per L-row. Grid = B*NSPLIT = 1024 blocks -> 4096 waves to saturate HBM.
__global__ void k_stats_partial(const float* __restrict__ residual,
                                float* __restrict__ part) {
  int t   = threadIdx.x;                 // 0..127
  int blk = blockIdx.x;                  // 0..B*NSPLIT-1
  int b   = blk / NSPLIT;
  int s   = blk % NSPLIT;
  const float4* p =
      (const float4*)(residual + ((size_t)b * LL + (size_t)s * LCHUNK) * CC) + t;
  float sa0 = 0.f, sa1 = 0.f, sa2 = 0.f, sa3 = 0.f;
  float mx0 = 0.f, mx1 = 0.f, mx2 = 0.f, mx3 = 0.f;
  float ss0 = 0.f, ss1 = 0.f, ss2 = 0.f, ss3 = 0.f;
#pragma unroll 4
  for (int l = 0; l < LCHUNK; ++l) {
    float4 v = p[(size_t)l * (CC / 4)];
    float a0 = fabsf(v.x), a1 = fabsf(v.y), a2 = fabsf(v.z), a3 = fabsf(v.w);
    sa0 += a0; sa1 += a1; sa2 += a2; sa3 += a3;
    mx0 = fmaxf(mx0, a0); mx1 = fmaxf(mx1, a1);
    mx2 = fmaxf(mx2, a2); mx3 = fmaxf(mx3, a3);
    ss0 += v.x * v.x; ss1 += v.y * v.y;
    ss2 += v.z * v.z; ss3 += v.w * v.w;
  }
  float* o = part + ((size_t)blk * CC + (size_t)t * 4) * 3;  // 12 contiguous
  o[0] = sa0; o[1]  = mx0; o[2]  = ss0;
  o[3] = sa1; o[4]  = mx1; o[5]  = ss1;
  o[6] = sa2; o[7]  = mx2; o[8]  = ss2;
  o[9] = sa3; o[10] = mx3; o[11] = ss3;
}

// ------- Kernel B: finalize h0 + all tiny per-row linears (4->16, 16->16) ---
__global__ void k_stats_finalize(const float* __restrict__ part,
                                 const float* __restrict__ W_l1, const float* __restrict__ b_l1,
                                 const float* __restrict__ W_l2, const float* __restrict__ b_l2,
                                 const float* __restrict__ W_g,  const float* __restrict__ b_g,
                                 const float* __restrict__ W_q,  const float* __restrict__ b_q,
                                 const float* __restrict__ W_k,  const float* __restrict__ b_k,
                                 float* __restrict__ h0_out,
                                 float* __restrict__ h_local, float* __restrict__ qv,
                                 float* __restrict__ kv, float* __restrict__ hg) {
  int idx = blockIdx.x * blockDim.x + threadIdx.x;   // 0..B*C-1
  if (idx >= BB * CC) return;
  int b = idx / CC, c = idx % CC;
  float sa = 0.f, mx = 0.f, ss = 0.f;
  for (int s = 0; s < NSPLIT; ++s) {
    const float* p = part + (((size_t)b * NSPLIT + s) * CC + c) * 3;
    sa += p[0];
    mx = fmaxf(mx, p[1]);
    ss += p[2];
  }
  float m  = sa * (1.0f / LL);
  float sq = ss * (1.0f / LL);
  float sd = sqrtf(fmaxf(sq - m * m, 0.0f));
  float h0[4] = {m, mx, sd, sq};
  float* ho = h0_out + (size_t)idx * 4;
  ho[0] = m; ho[1] = mx; ho[2] = sd; ho[3] = sq;

  float t1[HID];
#pragma unroll
  for (int i = 0; i < HID; ++i) {
    float acc = b_l1[i];
    for (int j = 0; j < 4; ++j) acc += W_l1[i * 4 + j] * h0[j];
    t1[i] = fmaxf(acc, 0.f);
  }
#pragma unroll
  for (int i = 0; i < HID; ++i) {
    float acc = b_l2[i];
    for (int j = 0; j < HID; ++j) acc += W_l2[i * HID + j] * t1[j];
    h_local[(size_t)idx * HID + i] = acc;
  }
#pragma unroll
  for (int i = 0; i < HID; ++i) {
    float acc = b_q[i];
    for (int j = 0; j < 4; ++j) acc += W_q[i * 4 + j] * h0[j];
    qv[(size_t)idx * HID + i] = acc;
  }
#pragma unroll
  for (int i = 0; i < HID; ++i) {
    float acc = b_k[i];
    for (int j = 0; j < 4; ++j) acc += W_k[i * 4 + j] * h0[j];
    kv[(size_t)idx * HID + i] = acc;
  }
#pragma unroll
  for (int i = 0; i < HID; ++i) {
    float acc = b_g[i];
    for (int j = 0; j < 4; ++j) acc += W_g[i * 4 + j] * h0[j];
    hg[(size_t)idx * HID + i] = acc;
  }
}

// ----------- Kernel C: sim_prior = relu(E @ E^T)  [512x512, K=16] -----------
__global__ void k_prior(const float* __restrict__ E, float* __restrict__ prior) {
  int lane = threadIdx.x & 31;
  int w    = (blockIdx.x * blockDim.x + threadIdx.x) >> 5;  // wave id, 0..1023
  int mt = w >> 5, nt = w & 31;
  int half = lane >> 4, lm = lane & 15;
  int row = mt * 16 + lm;
  int col = nt * 16 + lm;
  v8f acc = {};
#pragma unroll
  for (int k0 = 0; k0 < 16; k0 += 4) {
    int klo = k0 + half * 2;
    v2f a, bb;
    a.x  = E[row * 16 + klo];  a.y  = E[row * 16 + klo + 1];
    bb.x = E[col * 16 + klo];  bb.y = E[col * 16 + klo + 1];
    acc = wmma4(a, bb, acc);
  }
#pragma unroll
  for (int i = 0; i < 8; ++i) {
    int r  = mt * 16 + i + half * 8;
    int cc = nt * 16 + lm;
    prior[(size_t)r * CC + cc] = fmaxf(acc[i], 0.f);
  }
}

// --- Kernel D1: sim = relu(q k^T) + 0.2*prior + I  -> d_out A region --------
__global__ void k_sim(const float* __restrict__ qv, const float* __restrict__ kv,
                      const float* __restrict__ prior, float* __restrict__ sim) {
  int lane = threadIdx.x & 31;
  int w    = (blockIdx.x * blockDim.x + threadIdx.x) >> 5;   // 0..8191
  int b = w >> 10;
  int t = w & 1023;
  int mt = t >> 5, nt = t & 31;
  int half = lane >> 4, lm = lane & 15;
  int row = mt * 16 + lm;
  int col = nt * 16 + lm;
  const float* qb = qv + (size_t)b * CC * HID;
  const float* kb = kv + (size_t)b * CC * HID;
  v8f acc = {};
#pragma unroll
  for (int k0 = 0; k0 < 16; k0 += 4) {
    int klo = k0 + half * 2;
    v2f a, bb;
    a.x  = qb[row * HID + klo];  a.y  = qb[row * HID + klo + 1];
    bb.x = kb[col * HID + klo];  bb.y = kb[col * HID + klo + 1];
    acc = wmma4(a, bb, acc);
  }
  float* sb = sim + (size_t)b * CC * CC;
#pragma unroll
  for (int i = 0; i < 8; ++i) {
    int r  = mt * 16 + i + half * 8;
    int cc = nt * 16 + lm;
    float v = fmaxf(acc[i], 0.f) + 0.2f * prior[(size_t)r * CC + cc] +
              ((r == cc) ? 1.0f : 0.0f);
    sb[(size_t)r * CC + cc] = v;
  }
}

// --------------- Kernel D2: row softmax in place over last dim --------------
__global__ void k_softmax(float* __restrict__ A) {
  __shared__ float red[256];
  int r = blockIdx.x;                 // 0..B*C-1
  float* p = A + (size_t)r * CC;
  int t = threadIdx.x;
  float v0 = p[t], v1 = p[t + 256];
  red[t] = fmaxf(v0, v1);
  __syncthreads();
  for (int s = 128; s > 0; s >>= 1) {
    if (t < s) red[t] = fmaxf(red[t], red[t + s]);
    __syncthreads();
  }
  float mx = red[0];
  __syncthreads();
  float e0 = expf(v0 - mx), e1 = expf(v1 - mx);
  red[t] = e0 + e1;
  __syncthreads();
  for (int s = 128; s > 0; s >>= 1) {
    if (t < s) red[t] += red[t + s];
    __syncthreads();
  }
  float inv = 1.0f / red[0];
  p[t] = e0 * inv;
  p[t + 256] = e1 * inv;
}

// ------------- Kernel E: msg = A @ h_graph_in  [512x16, K=512] --------------
__global__ void k_msg(const float* __restrict__ A, const float* __restrict__ hg,
                      float* __restrict__ msg) {
  int lane = threadIdx.x & 31;
  int w    = (blockIdx.x * blockDim.x + threadIdx.x) >> 5;  // 0..255
  int b = w >> 5, mt = w & 31;
  int half = lane >> 4, lm = lane & 15;
  int row = mt * 16 + lm;
  const float* Ab = A  + (size_t)b * CC * CC;
  const float* hb = hg + (size_t)b * CC * HID;
  v8f acc = {};
  for (int k0 = 0; k0 < CC; k0 += 4) {
    int klo = k0 + half * 2;
    v2f a, bb;
    a.x  = Ab[(size_t)row * CC + klo];
    a.y  = Ab[(size_t)row * CC + klo + 1];
    bb.x = hb[(size_t)klo * HID + lm];
    bb.y = hb[(size_t)(klo + 1) * HID + lm];
    acc = wmma4(a, bb, acc);
  }
#pragma unroll
  for (int i = 0; i < 8; ++i) {
    int r = mt * 16 + i + half * 8;
    msg[((size_t)b * CC + r) * HID + lm] = acc[i];
  }
}

// ---- Kernel F: fusion + layernorm + head -> alpha (per (b,c) thread) -------
__global__ void k_head(const float* __restrict__ msg, const float* __restrict__ hg,
                       const float* __restrict__ h_local,
                       const float* __restrict__ W_f, const float* __restrict__ b_f,
                       const float* __restrict__ W_o, const float* __restrict__ b_o,
                       const float* __restrict__ ln_g, const float* __restrict__ ln_b,
                       float* __restrict__ alpha) {
  int idx = blockIdx.x * blockDim.x + threadIdx.x;   // 0..B*C-1
  if (idx >= BB * CC) return;
  float h[HID];
  float mu = 0.f;
#pragma unroll
  for (int i = 0; i < HID; ++i) {
    float hgr = hg[(size_t)idx * HID + i] +
                0.3f * fmaxf(msg[(size_t)idx * HID + i], 0.f);
    float v = h_local[(size_t)idx * HID + i] + 0.1f * hgr;
    h[i] = v;
    mu += v;
  }
  mu *= (1.0f / HID);
  float var = 0.f;
#pragma unroll
  for (int i = 0; i < HID; ++i) {
    float d = h[i] - mu;
    var += d * d;
  }
  var *= (1.0f / HID);
  float rstd = 1.0f / sqrtf(var + 1e-5f);
  float hn[HID];
#pragma unroll
  for (int i = 0; i < HID; ++i)
    hn[i] = (h[i] - mu) * rstd * ln_g[i] + ln_b[i];
  float h2[HID];
#pragma unroll
  for (int i = 0; i < HID; ++i) {
    float acc = b_f[i];
    for (int j = 0; j < HID; ++j) acc += W_f[i * HID + j] * hn[j];
    h2[i] = fmaxf(acc, 0.f);
  }
  float o = b_o[0];
#pragma unroll
  for (int j = 0; j < HID; ++j) o += W_o[j] * h2[j];
  alpha[idx] = 1.2f + 0.3f * tanhf(o);   // [B,1,C] flat == [b*C + c]
}

extern "C" void kernel_launch(void* const* d_in, const int* in_sizes, int n_in,
                              void* d_out, int out_size, void* d_ws, size_t ws_size,
                              hipStream_t stream) {
  (void)in_sizes; (void)n_in; (void)out_size; (void)ws_size;
  const float* residual = (const float*)d_in[0];
  const float* node_emb = (const float*)d_in[1];
  const float* W_l1 = (const float*)d_in[2];
  const float* b_l1 = (const float*)d_in[3];
  const float* W_l2 = (const float*)d_in[4];
  const float* b_l2 = (const float*)d_in[5];
  const float* W_g  = (const float*)d_in[6];
  const float* b_g  = (const float*)d_in[7];
  const float* W_q  = (const float*)d_in[8];
  const float* b_q  = (const float*)d_in[9];
  const float* W_k  = (const float*)d_in[10];
  const float* b_k  = (const float*)d_in[11];
  const float* W_f  = (const float*)d_in[12];
  const float* b_f  = (const float*)d_in[13];
  const float* W_o  = (const float*)d_in[14];
  const float* b_o  = (const float*)d_in[15];
  const float* ln_g = (const float*)d_in[16];
  const float* ln_b = (const float*)d_in[17];

  float* out   = (float*)d_out;
  float* alpha = out;                          // B*C        = 4096
  float* h0    = out + BB * CC;                // B*C*4      = 16384
  float* Aout  = out + BB * CC + BB * CC * 4;  // B*C*C      = 2097152

  float* ws      = (float*)d_ws;
  float* part    = ws + WS_PART;
  float* h_local = ws + WS_HLOCAL;
  float* qv      = ws + WS_Q;
  float* kv      = ws + WS_K;
  float* hgv     = ws + WS_HG;
  float* prior   = ws + WS_PRIOR;
  float* msg     = ws + WS_MSG;

  // 1) stats split-K over L (134 MB read; b128 loads, 4096 waves)
  k_stats_partial<<<BB * NSPLIT, 128, 0, stream>>>(residual, part);
  // 2) finalize stats + tiny per-row linears
  k_stats_finalize<<<(BB * CC + 255) / 256, 256, 0, stream>>>(
      part, W_l1, b_l1, W_l2, b_l2, W_g, b_g, W_q, b_q, W_k, b_k,
      h0, h_local, qv, kv, hgv);
  // 3) sim_prior = relu(E E^T) via fp32 WMMA
  k_prior<<<(CC / 16) * (CC / 16) / 8, 256, 0, stream>>>(node_emb, prior);
  // 4) sim = relu(q k^T) + 0.2*prior + I via fp32 WMMA
  k_sim<<<BB * (CC / 16) * (CC / 16) / 8, 256, 0, stream>>>(qv, kv, prior, Aout);
  // 5) row softmax -> A (in d_out)
  k_softmax<<<BB * CC, 256, 0, stream>>>(Aout);
  // 6) msg = A @ h_graph_in via fp32 WMMA (K=512)
  k_msg<<<BB * (CC / 16) / 8, 256, 0, stream>>>(Aout, hgv, msg);
  // 7) fusion + LN + head -> alpha
  k_head<<<(BB * CC + 255) / 256, 256, 0, stream>>>(
      msg, hgv, h_local, W_f, b_f, W_o, b_o, ln_g, ln_b, alpha);
}